// _LightGCN_80023830659227
// MI455X (gfx1250) — compile-verified
//
#include <hip/hip_runtime.h>
#include <stdint.h>
#include <math.h>

// ---------------------------------------------------------------------------
// LightGCN 2-layer propagation for MI455X (gfx1250, wave32).
//   x' = D^-1/2 A D^-1/2 x  done as:  acc[r] += dinv[c]*x[c];  x' = dinv[r]*acc
//   out = (x0 + x1 + x2) / 3
// Edge-index staging uses the CDNA5 Tensor Data Mover (tensor_load_to_lds,
// TENSORcnt) to DMA index chunks into LDS; row gathers are lane-striped
// global_load_b128; scatter uses native no-return global_atomic_add_f32.
// Streaming outputs use non-temporal hints to protect the L2-resident
// feature/accumulator working set (~307 MB vs 192 MB L2).
// ---------------------------------------------------------------------------

#define DIMV 128          // embedding dim
#define LANES 32          // wave32
#define F4_PER_ROW (DIMV / 4)
#define SCATTER_WG 256    // 8 waves
#define SCATTER_WAVES (SCATTER_WG / 32)
#define CHUNK 1024        // edges staged per workgroup via TDM

typedef float    v4f      __attribute__((ext_vector_type(4)));   // native vector (NT-capable)
typedef uint32_t tdm_g0_t __attribute__((ext_vector_type(4)));
typedef int      tdm_g1_t __attribute__((ext_vector_type(8)));
typedef int      tdm_g2_t __attribute__((ext_vector_type(4)));
typedef int      tdm_g4_t __attribute__((ext_vector_type(8)));

// ---- TDM 1-D tile load: tensor of 4-byte elements, tile of `tile_elems` ----
__device__ __forceinline__ void tdm_load_1d_i32(void* lds_dst, const int* gsrc,
                                                int tile_elems, long tensor_elems) {
  uint32_t lds_addr = (uint32_t)(uintptr_t)lds_dst;  // flat shared ptr: low 32 bits = LDS offset
  uint64_t ga = (uint64_t)(uintptr_t)gsrc;
  uint32_t td0 = (uint32_t)(tensor_elems > 0x7FFFFFFFL ? 0x7FFFFFFFL : tensor_elems);

  tdm_g0_t g0;
  g0[0] = 1u;                                        // count=1, user descriptor
  g0[1] = lds_addr;                                  // lds_addr [63:32]
  g0[2] = (uint32_t)(ga & 0xFFFFFFFFu);              // global_addr[31:0]
  g0[3] = (uint32_t)((ga >> 32) & 0x01FFFFFFu)       // global_addr[56:32]
          | (2u << 30);                              // type = 2 ("image")

  tdm_g1_t g1;
  g1[0] = (int)(2u << 16);                           // wg_mask=0, data_size=2 (4 bytes)
  g1[1] = (int)((td0 & 0xFFFFu) << 16);              // tensor_dim0[15:0] @ bits 63:48
  g1[2] = (int)(((td0 >> 16) & 0xFFFFu)              // tensor_dim0[31:16]
          | (1u << 16));                             // tensor_dim1 = 1
  g1[3] = (int)(((uint32_t)tile_elems & 0xFFFFu) << 16); // tile_dim0 @ bits 127:112
  g1[4] = 0;                                         // tile_dim1=0, tile_dim2=0 (unused)
  g1[5] = (int)td0;                                  // tensor_dim0_stride[31:0] (unused, 1-D)
  g1[6] = 0;
  g1[7] = 0;

  tdm_g2_t g2 = {0, 0, 0, 0};
  tdm_g2_t g3 = {0, 0, 0, 0};
  tdm_g4_t g4 = {0, 0, 0, 0, 0, 0, 0, 0};
  __builtin_amdgcn_tensor_load_to_lds(g0, g1, g2, g3, g4, 0);
  // The builtin only sees an integer LDS address; tell the compiler memory
  // (incl. the __shared__ staging buffers) has been written so subsequent
  // ds_loads are not folded to undef.
  asm volatile("" ::: "memory");
}

__device__ __forceinline__ void afadd(float* p, float v) {
  (void)__hip_atomic_fetch_add(p, v, __ATOMIC_RELAXED, __HIP_MEMORY_SCOPE_AGENT);
}

// ---------------------------------------------------------------------------
__global__ void zero_f4_kernel(v4f* __restrict__ p, long n4) {
  long i = (long)blockIdx.x * blockDim.x + threadIdx.x;
  long stride = (long)gridDim.x * blockDim.x;
  v4f z = {0.f, 0.f, 0.f, 0.f};
  for (; i < n4; i += stride) p[i] = z;
}

// deg[v] = 2 * incident-edge-count (reference bincounts concat([r,c]))
__global__ void deg_kernel(const int* __restrict__ u_idx, const int* __restrict__ i_idx,
                           int* __restrict__ deg, int E, int n_u) {
  int t = blockIdx.x * blockDim.x + threadIdx.x;
  if (t < E) {
    atomicAdd(&deg[u_idx[t]], 2);
    atomicAdd(&deg[n_u + i_idx[t]], 2);
  }
}

__global__ void dinv_kernel(const int* __restrict__ deg, float* __restrict__ dinv, int n) {
  int t = blockIdx.x * blockDim.x + threadIdx.x;
  if (t < n) {
    float d = fmaxf((float)deg[t], 1e-10f);
    dinv[t] = 1.0f / sqrtf(d);   // clip(deg)^-0.5
  }
}

// One chunk of CHUNK undirected edges per workgroup.
// TDM stages index tiles into LDS; each wave owns one edge per iteration:
//   acc[u]       += dinv[n_u+i] * x_i[i]
//   acc[n_u + i] += dinv[u]     * x_u[u]
__global__ __launch_bounds__(SCATTER_WG)
void scatter_kernel(const int* __restrict__ u_idx, const int* __restrict__ i_idx,
                    const float* __restrict__ x_u, const float* __restrict__ x_i,
                    const float* __restrict__ dinv, float* __restrict__ acc,
                    int E, int n_u) {
  __shared__ int sh_u[CHUNK];
  __shared__ int sh_i[CHUNK];

  int chunk_base = blockIdx.x * CHUNK;
  if (chunk_base >= E) return;
  int cnt = E - chunk_base;
  if (cnt > CHUNK) cnt = CHUNK;

  int wave = __builtin_amdgcn_readfirstlane((int)(threadIdx.x >> 5));
  int lane = (int)(threadIdx.x & 31);

  if (wave == 0) {
    long remain = (long)E - (long)chunk_base;
    tdm_load_1d_i32(sh_u, u_idx + chunk_base, cnt, remain);
    tdm_load_1d_i32(sh_i, i_idx + chunk_base, cnt, remain);
    __builtin_amdgcn_s_wait_tensorcnt(0);
  }
  __syncthreads();
  asm volatile("" ::: "memory");   // LDS now holds DMA'd indices

  for (int el = wave; el < cnt; el += SCATTER_WAVES) {
    int u  = __builtin_amdgcn_readfirstlane(sh_u[el]);   // user row (scalar)
    int iv = __builtin_amdgcn_readfirstlane(sh_i[el]);   // item row (scalar, 0-based)

    float wu = dinv[u];
    float wi = dinv[n_u + iv];

    const v4f xu = *(const v4f*)(x_u + (size_t)u  * DIMV + lane * 4);
    const v4f xi = *(const v4f*)(x_i + (size_t)iv * DIMV + lane * 4);

    float* au = acc + (size_t)u * DIMV + lane * 4;
    float* ai = acc + ((size_t)n_u + (size_t)iv) * DIMV + lane * 4;

    afadd(au + 0, wi * xi.x);
    afadd(au + 1, wi * xi.y);
    afadd(au + 2, wi * xi.z);
    afadd(au + 3, wi * xi.w);

    afadd(ai + 0, wu * xu.x);
    afadd(ai + 1, wu * xu.y);
    afadd(ai + 2, wu * xu.z);
    afadd(ai + 3, wu * xu.w);
  }
}

// Layer-1 finalize: x1 = dinv*acc ; out = (x0 + x1) / 3
__global__ void finalize1_kernel(const float* __restrict__ x0_u, const float* __restrict__ x0_i,
                                 const float* __restrict__ acc, const float* __restrict__ dinv,
                                 float* __restrict__ x1, float* __restrict__ out,
                                 int n, int n_u) {
  long t = (long)blockIdx.x * blockDim.x + threadIdx.x;   // one v4f each
  long total = (long)n * F4_PER_ROW;
  long stride = (long)gridDim.x * blockDim.x;
  const float inv3 = 1.0f / 3.0f;
  for (; t < total; t += stride) {
    long row = t >> 5;                 // F4_PER_ROW == 32
    int  c4  = (int)(t & 31);
    float w = dinv[row];
    v4f a = __builtin_nontemporal_load(((const v4f*)acc) + t);  // last use of acc
    v4f v1 = w * a;
    ((v4f*)x1)[t] = v1;                // x1 is hot next layer: keep cached
    v4f v0 = (row < n_u)
        ? ((const v4f*)x0_u)[row * F4_PER_ROW + c4]
        : ((const v4f*)x0_i)[(row - n_u) * F4_PER_ROW + c4];
    v4f o = (v0 + v1) * inv3;
    __builtin_nontemporal_store(o, ((v4f*)out) + t);   // streamed, re-read much later
  }
}

// Layer-2 finalize: out += (dinv*acc) / 3
__global__ void finalize2_kernel(const float* __restrict__ acc, const float* __restrict__ dinv,
                                 float* __restrict__ out, int n) {
  long t = (long)blockIdx.x * blockDim.x + threadIdx.x;
  long total = (long)n * F4_PER_ROW;
  long stride = (long)gridDim.x * blockDim.x;
  const float inv3 = 1.0f / 3.0f;
  for (; t < total; t += stride) {
    long row = t >> 5;
    float w = dinv[row] * inv3;
    v4f a = __builtin_nontemporal_load(((const v4f*)acc) + t);  // last use
    v4f o = __builtin_nontemporal_load(((const v4f*)out) + t);  // last use
    v4f r = o + w * a;
    __builtin_nontemporal_store(r, ((v4f*)out) + t);
  }
}

// ---------------------------------------------------------------------------
extern "C" void kernel_launch(void* const* d_in, const int* in_sizes, int n_in,
                              void* d_out, int out_size, void* d_ws, size_t ws_size,
                              hipStream_t stream) {
  const float* u_emb = (const float*)d_in[0];
  const float* i_emb = (const float*)d_in[1];
  const int*   u_idx = (const int*)d_in[2];
  const int*   i_idx = (const int*)d_in[3];
  float* out = (float*)d_out;

  const int n_u = in_sizes[0] / DIMV;
  const int n_i = in_sizes[1] / DIMV;
  const int E   = in_sizes[2];
  const int n   = n_u + n_i;

  // workspace carve-up (256B aligned): deg | dinv | x1 | acc
  char* ws = (char*)d_ws;
  size_t off = 0;
  auto take = [&](size_t bytes) -> void* {
    void* p = (void*)(ws + off);
    off += (bytes + 255) & ~(size_t)255;
    return p;
  };
  int*   deg  = (int*)  take((size_t)n * sizeof(int));
  float* dinv = (float*)take((size_t)n * sizeof(float));
  float* x1   = (float*)take((size_t)n * DIMV * sizeof(float));
  float* acc  = (float*)take((size_t)n * DIMV * sizeof(float));
  (void)ws_size; (void)n_in; (void)out_size;

  const int ZBLK = 256;
  const long accN4 = (long)n * F4_PER_ROW;           // v4f count of acc / x1
  const long degN4 = ((long)n + 3) / 4;              // deg zeroed as v4f (padded alloc)
  const int zgridAcc = 4096;
  const int zgridDeg = (int)((degN4 + ZBLK - 1) / ZBLK);

  // degrees -> dinv
  zero_f4_kernel<<<zgridDeg, ZBLK, 0, stream>>>((v4f*)deg, degN4);
  deg_kernel<<<(E + 255) / 256, 256, 0, stream>>>(u_idx, i_idx, deg, E, n_u);
  dinv_kernel<<<(n + 255) / 256, 256, 0, stream>>>(deg, dinv, n);

  const int scatterBlocks = (E + CHUNK - 1) / CHUNK;

  // ---- layer 1: x1 = Prop(x0), out = (x0 + x1)/3 ----
  zero_f4_kernel<<<zgridAcc, ZBLK, 0, stream>>>((v4f*)acc, accN4);
  scatter_kernel<<<scatterBlocks, SCATTER_WG, 0, stream>>>(
      u_idx, i_idx, u_emb, i_emb, dinv, acc, E, n_u);
  finalize1_kernel<<<4096, 256, 0, stream>>>(u_emb, i_emb, acc, dinv, x1, out, n, n_u);

  // ---- layer 2: x2 = Prop(x1), out += x2/3 ----
  zero_f4_kernel<<<zgridAcc, ZBLK, 0, stream>>>((v4f*)acc, accN4);
  scatter_kernel<<<scatterBlocks, SCATTER_WG, 0, stream>>>(
      u_idx, i_idx, x1, x1 + (size_t)n_u * DIMV, dinv, acc, E, n_u);
  finalize2_kernel<<<4096, 256, 0, stream>>>(acc, dinv, out, n);
}